// GATLayer_34187939676735
// MI455X (gfx1250) — compile-verified
//
#include <hip/hip_runtime.h>
#include <hip/hip_bf16.h>

// GAT layer for MI455X (gfx1250, wave32).
// B=4, N=512, C_IN=64, H=4, D=32, TOTAL=128.
//
// Pipeline:
//   K1: q,k,v = h @ W{q,k,v}          -- V_WMMA_F32_16X16X4_F32 tiles
//   K2: s_q,s_k per-head dot products
//   K3: fused: TDM v-tile DMA to LDS (overlapped) || scores -> softmax,
//       then (alpha @ v) WMMA entirely from LDS, ELU, store.
//
// Workspace layout (floats):
//   qkv : [3][2048][128]   at 0
//   s   : [2][4][512][4]   at 3*2048*128   (s_q then s_k)

typedef __attribute__((ext_vector_type(2))) float v2f;
typedef __attribute__((ext_vector_type(8))) float v8f;
typedef __attribute__((ext_vector_type(4))) unsigned int u32x4;
typedef __attribute__((ext_vector_type(8))) int i32x8;
typedef __attribute__((ext_vector_type(4))) int i32x4;

#if __has_builtin(__builtin_amdgcn_tensor_load_to_lds) && \
    __has_builtin(__builtin_amdgcn_s_wait_tensorcnt)
#define USE_TDM 1
#else
#define USE_TDM 0
#endif

#define GB   4
#define GN   512
#define CIN  64
#define GH   4
#define GD   32
#define TOT  128
#define BN   (GB * GN)          // 2048
#define VROW 33                 // 32 floats + 1 pad (TDM pad_interval=32dw, pad=1dw)

// ---------------------------------------------------------------------------
// Kernel 1: QKV projection. One wave per 16x16 output tile.
// Tiles: 3 mats * (2048/16) Mtiles * (128/16) Ntiles = 3072 waves.
// ---------------------------------------------------------------------------
__global__ __launch_bounds__(256) void gat_qkv_wmma(
    const float* __restrict__ h,
    const float* __restrict__ Wq,
    const float* __restrict__ Wk,
    const float* __restrict__ Wv,
    float* __restrict__ qkv) {
  const int lane = threadIdx.x & 31;
  const int wid  = blockIdx.x * (blockDim.x >> 5) + (threadIdx.x >> 5);
  if (wid >= 3 * 128 * 8) return;          // wave-uniform guard

  const int mat = wid >> 10;               // 0=q 1=k 2=v
  const int rem = wid & 1023;
  const int m0  = (rem >> 3) * 16;
  const int n0  = (rem & 7) * 16;

  const float* W = (mat == 0) ? Wq : (mat == 1) ? Wk : Wv;
  float* out = qkv + (size_t)mat * BN * TOT;

  // A-fragment (16x4 f32): lane L -> row (L&15), K pair at 2*(L>>4)
  const int arow = m0 + (lane & 15);
  const int koff = (lane >> 4) * 2;
  const int bcol = n0 + (lane & 15);

  v8f acc = {};
  #pragma unroll
  for (int k = 0; k < CIN; k += 4) {
    v2f a, bf;
    a.x  = h[arow * CIN + k + koff];
    a.y  = h[arow * CIN + k + koff + 1];
    bf.x = W[(k + koff)     * TOT + bcol];
    bf.y = W[(k + koff + 1) * TOT + bcol];
    acc = __builtin_amdgcn_wmma_f32_16x16x4_f32(false, a, false, bf,
                                                (short)0, acc, false, false);
  }
  // C/D layout: VGPR r -> M = r (lanes 0-15) / r+8 (lanes 16-31), N = lane&15
  const int mbase = m0 + ((lane >> 4) << 3);
  #pragma unroll
  for (int r = 0; r < 8; ++r)
    out[(size_t)(mbase + r) * TOT + bcol] = acc[r];
}

// ---------------------------------------------------------------------------
// Kernel 2: s_q[b,n,h] = sum_d q[b,n,h,d]*a_q[h,d]; same for s_k.
// ---------------------------------------------------------------------------
__global__ __launch_bounds__(256) void gat_head_scores(
    const float* __restrict__ qkv,
    const float* __restrict__ a_q,
    const float* __restrict__ a_k,
    float* __restrict__ s) {
  const int t = blockIdx.x * blockDim.x + threadIdx.x;
  if (t >= 2 * BN * GH) return;
  const int which = t >> 13;               // 0: q/a_q, 1: k/a_k
  const int idx   = t & 8191;              // (b*N+n)*H + h
  const int hh    = idx & (GH - 1);
  const int bn    = idx >> 2;
  const float* src = qkv + (size_t)which * BN * TOT + (size_t)bn * TOT + hh * GD;
  const float* av  = (which == 0) ? a_q : a_k;
  float acc = 0.f;
  #pragma unroll
  for (int d = 0; d < GD; ++d) acc += src[d] * av[hh * GD + d];
  s[t] = acc;
}

// ---------------------------------------------------------------------------
// Kernel 3: fused attention. One workgroup per (b, 16-row tile, head).
//   TDM: v-tile (512 x 32 f32, row stride 128) -> LDS, padded to 33-f rows,
//        issued by wave 0 at entry, overlapped with phases A/B.
//   A: masked leaky-relu scores (16x512) -> LDS
//   B: row softmax (one wave32 per 2 rows)
//   C: out16x32 = alpha16x512 @ v512x32 via WMMA f32 16x16x4, all-LDS feeds
//      (8 waves: 2 N-tiles x 4 K-groups of 128)
//   D: reduce K-group partials, ELU, store
// ---------------------------------------------------------------------------
#define SCPAD 516   // 516 % 64 == 4 -> conflict-free column gathers
__global__ __launch_bounds__(256) void gat_attn_aggregate(
    const int*   __restrict__ adj,
    const float* __restrict__ edge_feat,   // (B,N,N,2)
    const float* __restrict__ We,          // (2,128)
    const float* __restrict__ a_e,         // (4,32)
    const float* __restrict__ qkv,
    const float* __restrict__ s,
    float* __restrict__ outp) {
  __shared__ float sc[16][SCPAD];
  __shared__ float part[4][16][33];
  __shared__ float vtile[GN * VROW];       // 512 x (32+1) floats = 66 KB

  const int tid = threadIdx.x;
  const int blk = blockIdx.x;              // b*(32*4) + it*4 + h
  const int hh  = blk & 3;
  const int it  = (blk >> 2) & 31;
  const int b   = blk >> 7;
  const int i0  = it * 16;

  const float* vsrc = qkv + (size_t)2 * BN * TOT + (size_t)b * GN * TOT + hh * GD;

#if USE_TDM
  // --- Tensor Data Mover: one DMA per workgroup, issued by wave 0 only ---
  if (tid < 32) {
    const unsigned lds_off = (unsigned)(uintptr_t)(void*)vtile;  // LDS byte offset
    const unsigned long long ga = (unsigned long long)(uintptr_t)vsrc;
    // D# group 0: count=1 | lds_addr | global_addr[56:0] | type=2
    u32x4 g0 = { 1u,
                 lds_off,
                 (unsigned)(ga & 0xffffffffu),
                 (unsigned)(((ga >> 32) & 0x01ffffffu) | (2u << 30)) };
    // D# group 1: data_size=4B, pad_enable, pad_interval=32dw, pad_amount=1dw,
    //             tensor_dim0=32, tensor_dim1=512, tile_dim0=32, tile_dim1=512,
    //             tensor_dim0_stride=128
    i32x8 g1 = { (int)((2u << 16) | (1u << 20) | (4u << 22)),
                 (int)(32u << 16),   // tensor_dim0[15:0] in [31:16]
                 (int)(512u << 16),  // tensor_dim0[31:16]=0 | tensor_dim1[15:0]
                 (int)(32u << 16),   // tensor_dim1[31:16]=0 | tile_dim0=32
                 (int)512,           // tile_dim1=512 | tile_dim2=0
                 (int)TOT,           // tensor_dim0_stride[31:0] = 128
                 0, 0 };
    i32x4 gz4 = {};
    i32x8 gz8 = {};
    __builtin_amdgcn_tensor_load_to_lds(g0, g1, gz4, gz4, gz8, 0);
  }
#endif

  // Collapsed edge-feature coefficients (redundant per thread; 64 FMAs)
  float c0 = 0.f, c1 = 0.f;
  #pragma unroll
  for (int d = 0; d < GD; ++d) {
    const float ae = a_e[hh * GD + d];
    c0 += We[hh * GD + d]       * ae;
    c1 += We[TOT + hh * GD + d] * ae;
  }

  const float* s_q = s;
  const float* s_k = s + BN * GH;
  const float2* ef2 = (const float2*)edge_feat + (size_t)b * GN * GN;

  // Phase A: scores
  for (int idx = tid; idx < 16 * GN; idx += 256) {
    const int il = idx >> 9;
    const int j  = idx & (GN - 1);
    const int i  = i0 + il;
    const float2 e2 = ef2[(size_t)i * GN + j];
    float raw = s_q[(b * GN + i) * GH + hh] + s_k[(b * GN + j) * GH + hh]
              + e2.x * c0 + e2.y * c1;
    raw = (raw > 0.f) ? raw : 0.2f * raw;             // leaky_relu(0.2)
    sc[il][j] = (adj[i * GN + j] > 0) ? raw : -1e9f;  // log(1)=0 when adj==1
  }

#if !USE_TDM
  // Fallback: cooperative staged copy of the v tile into padded LDS rows.
  for (int idx = tid; idx < GN * 8; idx += 256) {
    const int r = idx >> 3, c4 = (idx & 7) * 4;
    const float4 t = *(const float4*)(vsrc + (size_t)r * TOT + c4);
    vtile[r * VROW + c4 + 0] = t.x;
    vtile[r * VROW + c4 + 1] = t.y;
    vtile[r * VROW + c4 + 2] = t.z;
    vtile[r * VROW + c4 + 3] = t.w;
  }
#endif
  __syncthreads();

  // Phase B: softmax over j (wave w owns rows 2w, 2w+1)
  const int w = tid >> 5, lane = tid & 31;
  #pragma unroll
  for (int rr = 0; rr < 2; ++rr) {
    const int row = w * 2 + rr;
    float m = -3.0e38f;
    for (int j = lane; j < GN; j += 32) m = fmaxf(m, sc[row][j]);
    #pragma unroll
    for (int off = 16; off > 0; off >>= 1) m = fmaxf(m, __shfl_xor(m, off, 32));
    float sum = 0.f;
    for (int j = lane; j < GN; j += 32) {
      const float e = __expf(sc[row][j] - m);
      sc[row][j] = e;
      sum += e;
    }
    #pragma unroll
    for (int off = 16; off > 0; off >>= 1) sum += __shfl_xor(sum, off, 32);
    const float inv = 1.f / sum;
    for (int j = lane; j < GN; j += 32) sc[row][j] *= inv;
  }

#if USE_TDM
  __builtin_amdgcn_s_wait_tensorcnt(0);    // wave 0's DMA done (others trivially 0)
#endif
  __syncthreads();                         // publish softmax + v tile to all waves

  // Phase C: WMMA aggregation, A and B fragments both from LDS
  const int ntile = w & 1;                 // D halves 0..15 / 16..31
  const int kg    = w >> 1;                // K group of 128
  const int arow  = lane & 15;
  const int koff  = (lane >> 4) * 2;
  const int ncol  = (lane & 15) + ntile * 16;

  v8f acc = {};
  #pragma unroll 4
  for (int ks = 0; ks < 32; ++ks) {
    const int kb = kg * 128 + ks * 4;
    v2f a, bf;
    a.x  = sc[arow][kb + koff];
    a.y  = sc[arow][kb + koff + 1];
    bf.x = vtile[(kb + koff)     * VROW + ncol];
    bf.y = vtile[(kb + koff + 1) * VROW + ncol];
    acc = __builtin_amdgcn_wmma_f32_16x16x4_f32(false, a, false, bf,
                                                (short)0, acc, false, false);
  }
  const int mb = (lane >> 4) << 3;
  #pragma unroll
  for (int r = 0; r < 8; ++r)
    part[kg][mb + r][ncol] = acc[r];
  __syncthreads();

  // Phase D: reduce partials + ELU + store
  for (int o = tid; o < 16 * GD; o += 256) {
    const int m = o >> 5, d = o & 31;
    float val = part[0][m][d] + part[1][m][d] + part[2][m][d] + part[3][m][d];
    val = (val > 0.f) ? val : (__expf(val) - 1.f);    // ELU
    outp[(size_t)(b * GN + i0 + m) * TOT + hh * GD + d] = val;
  }
}

// ---------------------------------------------------------------------------
extern "C" void kernel_launch(void* const* d_in, const int* in_sizes, int n_in,
                              void* d_out, int out_size, void* d_ws, size_t ws_size,
                              hipStream_t stream) {
  const float* h         = (const float*)d_in[0];
  const int*   adj       = (const int*)  d_in[1];
  const float* edge_feat = (const float*)d_in[2];
  const float* Wq        = (const float*)d_in[3];
  const float* Wk        = (const float*)d_in[4];
  const float* Wv        = (const float*)d_in[5];
  const float* We        = (const float*)d_in[6];
  const float* a_q       = (const float*)d_in[7];
  const float* a_k       = (const float*)d_in[8];
  const float* a_e       = (const float*)d_in[9];
  float* out = (float*)d_out;

  float* ws  = (float*)d_ws;
  float* qkv = ws;                         // 3*2048*128 = 786432 floats
  float* s   = ws + 3 * BN * TOT;          // 2*8192 floats

  // K1: 3072 wave-tiles, 8 waves/block
  gat_qkv_wmma<<<384, 256, 0, stream>>>(h, Wq, Wk, Wv, qkv);
  // K2: 16384 threads
  gat_head_scores<<<64, 256, 0, stream>>>(qkv, a_q, a_k, s);
  // K3: B * (N/16) * H = 512 workgroups
  gat_attn_aggregate<<<512, 256, 0, stream>>>(adj, edge_feat, We, a_e, qkv, s, out);
}